// ControlNorm1DLoop_89696097010204
// MI455X (gfx1250) — compile-verified
//
#include <hip/hip_runtime.h>
#include <stdint.h>

// Online-normalization scan, chunked-transfer-function formulation.
//   K1: per-chunk zero-state stats (B, sumE, sumE2)   [reads x via TDM->LDS]
//   K2: exact 64-step scan of chunk transfer maps -> (muIn, vIn) per chunk
//   K3: per-chunk exact recurrence replay, writes out [reads x via TDM->LDS]

#define AFWD_F 0.999f
#define EPS_F  1e-5f

constexpr int COLS      = 256;   // columns per block == blockDim.x
constexpr int TILE_ROWS = 16;    // rows per TDM tile (16*256*4B = 16 KiB/buffer)
constexpr int CHUNK     = 128;   // rows per chunk

#if defined(__gfx1250__) && __has_builtin(__builtin_amdgcn_tensor_load_to_lds) && __has_builtin(__builtin_amdgcn_s_wait_tensorcnt)
#define USE_TDM 1
#else
#define USE_TDM 0
#endif

typedef unsigned int u32x4 __attribute__((ext_vector_type(4)));
typedef int          i32x8 __attribute__((ext_vector_type(8)));
typedef int          i32x4 __attribute__((ext_vector_type(4)));

#if USE_TDM
// Build a 2D tensor-DMA descriptor (D#) and issue TENSOR_LOAD_TO_LDS.
// tile = rows x cols fp32, row stride = stride_elems, dest = LDS byte offset.
__device__ __forceinline__ void tdm_issue_tile(const float* gsrc, unsigned ldsoff,
                                               int rows, int cols, int stride_elems) {
  uint64_t ga = (uint64_t)(uintptr_t)gsrc;
  u32x4 g0;
  g0.x = 1u;                                            // count=1, user mode
  g0.y = ldsoff;                                        // lds_addr [63:32]
  g0.z = (unsigned)(ga & 0xFFFFFFFFull);                // global_addr lo
  g0.w = (unsigned)((ga >> 32) & 0x1FFFFFFull)          // global_addr[56:32]
       | (2u << 30);                                    // type=2 ("image")
  unsigned td0 = (unsigned)cols, td1 = (unsigned)rows;
  i32x8 g1;
  g1[0] = (int)(2u << 16);                              // data_size=2 (4 bytes), mask=0
  g1[1] = (int)((td0 & 0xFFFFu) << 16);                 // tensor_dim0[15:0] @ bits[63:48]
  g1[2] = (int)(((td0 >> 16) & 0xFFFFu)                 // tensor_dim0[31:16]
       | ((td1 & 0xFFFFu) << 16));                      // tensor_dim1[15:0]
  g1[3] = (int)(((td1 >> 16) & 0xFFFFu)                 // tensor_dim1[31:16]
       | ((td0 & 0xFFFFu) << 16));                      // tile_dim0 = cols
  g1[4] = (int)(td1 & 0xFFFFu);                         // tile_dim1 = rows, tile_dim2=0
  g1[5] = (int)(unsigned)stride_elems;                  // tensor_dim0_stride lo32
  g1[6] = 0;                                            // stride0 hi16, stride1 lo16
  g1[7] = 0;
  i32x4 z4 = {0, 0, 0, 0};
  i32x8 z8 = {0, 0, 0, 0, 0, 0, 0, 0};
  __builtin_amdgcn_tensor_load_to_lds(g0, g1, z4, z4, z8, 0);
}
#endif

__device__ __forceinline__ void load_tile(float* sbuf, const float* gsrc, int L) {
#if USE_TDM
  if (threadIdx.x < 32u) {
    tdm_issue_tile(gsrc, (unsigned)(uintptr_t)sbuf, TILE_ROWS, COLS, L);
  }
#else
  for (int i = (int)threadIdx.x; i < TILE_ROWS * COLS; i += (int)blockDim.x) {
    int r = i / COLS, c = i % COLS;
    sbuf[i] = gsrc[(size_t)r * L + c];
  }
#endif
}

__device__ __forceinline__ void wait_tiles_le1() {
#if USE_TDM
  if (threadIdx.x < 32u) __builtin_amdgcn_s_wait_tensorcnt(1);
#endif
}
__device__ __forceinline__ void wait_tiles_0() {
#if USE_TDM
  if (threadIdx.x < 32u) __builtin_amdgcn_s_wait_tensorcnt(0);
#endif
}

// ---------------- K1: per-chunk zero-state statistics --------------------
__global__ void __launch_bounds__(COLS)
k_chunk_stats(const float* __restrict__ x, float* __restrict__ B,
              float* __restrict__ sE, float* __restrict__ sE2, int L) {
  __shared__ float sm[2][TILE_ROWS * COLS];
  const int cg = blockIdx.x, g = blockIdx.y;
  const int col = cg * COLS + (int)threadIdx.x;
  const float* base = x + (size_t)g * CHUNK * L + (size_t)cg * COLS;
  constexpr int NT = CHUNK / TILE_ROWS;

  load_tile(sm[0], base, L);

  float c = 0.f, se = 0.f, se2 = 0.f;
  const float a = AFWD_F, oma = 1.0f - AFWD_F;
  for (int t = 0; t < NT; ++t) {
    if (t + 1 < NT) {
      load_tile(sm[(t + 1) & 1], base + (size_t)(t + 1) * TILE_ROWS * L, L);
      wait_tiles_le1();            // current tile (issued earlier) complete
    } else {
      wait_tiles_0();
    }
    __syncthreads();
    const float* buf = sm[t & 1];
#pragma unroll
    for (int r = 0; r < TILE_ROWS; ++r) {
      float xv = buf[r * COLS + threadIdx.x];
      float e = xv - c;            // e_t = x_t - c_t (pre-update)
      se += e;                     // Σ e_t
      se2 = a * se2 + e * e;       // Σ a^{C-1-t} e_t^2
      c = a * c + oma * xv;        // c_{t+1}
    }
    __syncthreads();
  }
  size_t idx = (size_t)g * L + col;
  B[idx] = c;  sE[idx] = se;  sE2[idx] = se2;
}

// ---------------- K2: exact scan over chunk transfer maps ----------------
__global__ void k_chunk_scan(const float* __restrict__ m0, const float* __restrict__ v0,
                             const float* __restrict__ B, const float* __restrict__ sE,
                             const float* __restrict__ sE2,
                             float* __restrict__ muIn, float* __restrict__ vIn,
                             int L, int G) {
  int l = (int)(blockIdx.x * blockDim.x + threadIdx.x);
  if (l >= L) return;
  // aC = AFWD^CHUNK by repeated squaring (deterministic)
  float aC = 1.0f;
  { float p = AFWD_F; int n = CHUNK; while (n) { if (n & 1) aC *= p; p *= p; n >>= 1; } }
  const float R  = aC * (1.0f - aC);             // μ² coefficient (constant)
  const float kp = AFWD_F * (1.0f - AFWD_F);     // P = kp * sumE2
  const float kq = -2.0f * (1.0f - AFWD_F) * aC; // Q = kq * sumE
  float mu = m0[l], v = v0[l];
  for (int g = 0; g < G; ++g) {
    size_t idx = (size_t)g * L + l;
    muIn[idx] = mu;  vIn[idx] = v;
    float b = B[idx], se = sE[idx], s2 = sE2[idx];
    v  = aC * v + kp * s2 + kq * se * mu + R * mu * mu;
    mu = aC * mu + b;
  }
}

// ---------------- K3: replay recurrence with exact boundary state --------
__global__ void __launch_bounds__(COLS)
k_finalize(const float* __restrict__ x, const float* __restrict__ muIn,
           const float* __restrict__ vIn, float* __restrict__ out, int L) {
  __shared__ float sm[2][TILE_ROWS * COLS];
  const int cg = blockIdx.x, g = blockIdx.y;
  const int col = cg * COLS + (int)threadIdx.x;
  const float* base = x + (size_t)g * CHUNK * L + (size_t)cg * COLS;
  float* obase = out + (size_t)g * CHUNK * L + (size_t)cg * COLS;
  constexpr int NT = CHUNK / TILE_ROWS;

  load_tile(sm[0], base, L);

  float mu = muIn[(size_t)g * L + col];
  float v  = vIn[(size_t)g * L + col];
  const float a = AFWD_F, oma = 1.0f - AFWD_F, aoma = AFWD_F * (1.0f - AFWD_F);
  for (int t = 0; t < NT; ++t) {
    if (t + 1 < NT) {
      load_tile(sm[(t + 1) & 1], base + (size_t)(t + 1) * TILE_ROWS * L, L);
      wait_tiles_le1();
    } else {
      wait_tiles_0();
    }
    __syncthreads();
    const float* buf = sm[t & 1];
#pragma unroll
    for (int r = 0; r < TILE_ROWS; ++r) {
      float xv = buf[r * COLS + threadIdx.x];
      float d = xv - mu;
      float o = d * rsqrtf(v + EPS_F);           // uses v BEFORE update (matches ref)
      obase[(size_t)(t * TILE_ROWS + r) * L + threadIdx.x] = o;
      v = a * v + aoma * d * d;
      mu += oma * d;
    }
    __syncthreads();
  }
}

extern "C" void kernel_launch(void* const* d_in, const int* in_sizes, int n_in,
                              void* d_out, int out_size, void* d_ws, size_t ws_size,
                              hipStream_t stream) {
  const float* x   = (const float*)d_in[0];
  const float* m   = (const float*)d_in[1];
  const float* var = (const float*)d_in[2];
  float* out = (float*)d_out;

  const int L = in_sizes[1];
  const int N = in_sizes[0] / L;
  const int G = N / CHUNK;
  const size_t GL = (size_t)G * (size_t)L;

  float* ws   = (float*)d_ws;
  float* B    = ws;
  float* sE   = ws + GL;
  float* sE2  = ws + 2 * GL;
  float* muIn = ws + 3 * GL;
  float* vIn  = ws + 4 * GL;

  dim3 grid((unsigned)(L / COLS), (unsigned)G), blk((unsigned)COLS);
  k_chunk_stats<<<grid, blk, 0, stream>>>(x, B, sE, sE2, L);
  k_chunk_scan<<<(L + 255) / 256, 256, 0, stream>>>(m, var, B, sE, sE2, muIn, vIn, L, G);
  k_finalize<<<grid, blk, 0, stream>>>(x, muIn, vIn, out, L);
}